// GNN_18562848653544
// MI455X (gfx1250) — compile-verified
//
#include <hip/hip_runtime.h>
#include <hip/hip_bf16.h>

#define NN 100000
#define NE 3200000
#define INF 128
#define OUTF 64

typedef __attribute__((ext_vector_type(2))) float v2f;
typedef __attribute__((ext_vector_type(8))) float v8f;

// ---------------- degree init: deg[i] = 1.0 (self loop) ----------------
__global__ void k_init_deg(float* __restrict__ deg, int n) {
    int i = blockIdx.x * blockDim.x + threadIdx.x;
    if (i < n) deg[i] = 1.0f;
}

// ---------------- degree accumulation at targets ----------------
__global__ void k_accum_deg(const int* __restrict__ col, float* __restrict__ deg, int e_cnt) {
    int e = blockIdx.x * blockDim.x + threadIdx.x;
    if (e < e_cnt) atomicAdd(&deg[col[e]], 1.0f);
}

// ---------------- dinv = rsqrt(deg), in place ----------------
__global__ void k_rsqrt(float* __restrict__ deg, int n) {
    int i = blockIdx.x * blockDim.x + threadIdx.x;
    if (i < n) deg[i] = rsqrtf(deg[i]);   // deg >= 1 always (self loops)
}

// ---------------- h = x @ W via V_WMMA_F32_16X16X4_F32 ----------------
// One wave: 16 nodes x 64 out-features (4 v8f accumulators).
// W staged in LDS pair-interleaved: (W[k][n], W[k+1][n]) contiguous, with the
// odd k-pair's columns rotated by +16 so the two lane halves of a wave hit
// disjoint LDS bank halves -> each B fragment is one conflict-free ds_load_b64.
__global__ void __launch_bounds__(256)
k_gemm_wmma(const float* __restrict__ x, const float* __restrict__ W,
            float* __restrict__ h, int n) {
    __shared__ float Wlds[INF * OUTF];   // 32 KB

    // cooperative pair-interleaved W staging
    // layout: j = k>>1 (k-pair), q = j>>1, p = j&1
    //   Wlds[q*256 + p*128 + (((n + 16p) & 63) << 1) + {0,1}] = W[2j..2j+1][n]
    for (int t = threadIdx.x; t < (INF / 2) * OUTF; t += blockDim.x) {
        int j = t >> 6;            // k-pair 0..63
        int c = t & 63;            // column
        int q = j >> 1, p = j & 1;
        v2f w;
        w.x = W[(2 * j    ) * OUTF + c];
        w.y = W[(2 * j + 1) * OUTF + c];
        int d = q * 256 + p * 128 + (((c + (p << 4)) & 63) << 1);
        *(v2f*)(Wlds + d) = w;
    }
    __syncthreads();

    const int lane  = threadIdx.x & 31;
    const int wave  = blockIdx.x * (blockDim.x >> 5) + (threadIdx.x >> 5);
    const int node0 = wave * 16;
    if (node0 >= n) return;              // wave-uniform: EXEC stays all-1s

    const int m = lane & 15;             // A row / B,C column within tile
    const int p = lane >> 4;             // 0/1: which k-pair of the quad
    const int rot = p << 4;

    // loop-invariant per-lane LDS addresses for the 4 ntiles
    const float* wb0 = Wlds + p * 128 + (((m +  0 + rot) & 63) << 1);
    const float* wb1 = Wlds + p * 128 + (((m + 16 + rot) & 63) << 1);
    const float* wb2 = Wlds + p * 128 + (((m + 32 + rot) & 63) << 1);
    const float* wb3 = Wlds + p * 128 + (((m + 48 + rot) & 63) << 1);

    v8f acc0 = {}, acc1 = {}, acc2 = {}, acc3 = {};
    const float* xrow = x + (size_t)(node0 + m) * INF + (p << 1);

    #pragma unroll
    for (int k0 = 0; k0 < INF; k0 += 4) {
        // A fragment: x[node0+m][k0 + 2p .. +1]  (16x4 f32 A layout)
        v2f a = *(const v2f*)(xrow + k0);
        const int qoff = (k0 >> 2) * 256;   // folds into ds immediate offsets
        v2f b0 = *(const v2f*)(wb0 + qoff);
        v2f b1 = *(const v2f*)(wb1 + qoff);
        v2f b2 = *(const v2f*)(wb2 + qoff);
        v2f b3 = *(const v2f*)(wb3 + qoff);
        acc0 = __builtin_amdgcn_wmma_f32_16x16x4_f32(false, a, false, b0, (short)0, acc0, false, false);
        acc1 = __builtin_amdgcn_wmma_f32_16x16x4_f32(false, a, false, b1, (short)0, acc1, false, false);
        acc2 = __builtin_amdgcn_wmma_f32_16x16x4_f32(false, a, false, b2, (short)0, acc2, false, false);
        acc3 = __builtin_amdgcn_wmma_f32_16x16x4_f32(false, a, false, b3, (short)0, acc3, false, false);
    }

    // C/D layout: VGPR r -> row (r + 8*p), col = lane&15 (+ntile*16)
    float* hb = h + (size_t)(node0 + (p << 3)) * OUTF + m;
    #pragma unroll
    for (int r = 0; r < 8; ++r) {
        hb[r * OUTF +  0] = acc0[r];
        hb[r * OUTF + 16] = acc1[r];
        hb[r * OUTF + 32] = acc2[r];
        hb[r * OUTF + 48] = acc3[r];
    }
}

// ---------------- out = h * dinv^2  (self-loop message; initializes out) ----
__global__ void k_selfloop(const float* __restrict__ h, const float* __restrict__ dinv,
                           float* __restrict__ out, int n) {
    int idx = blockIdx.x * blockDim.x + threadIdx.x;   // n*16 threads, float4 each
    int i = idx >> 4;
    if (i >= n) return;
    int f4 = (idx & 15) << 2;
    float s = dinv[i]; s *= s;
    float4 v = *(const float4*)(h + (size_t)i * OUTF + f4);
    v.x *= s; v.y *= s; v.z *= s; v.w *= s;
    *(float4*)(out + (size_t)i * OUTF + f4) = v;
}

// ---------------- edge scatter: out[col] += h[row] * dinv[row]*dinv[col] ----
// 16 threads per edge, float4 gather, 4 scalar f32 atomics (no-return path).
__global__ void k_scatter(const int* __restrict__ row, const int* __restrict__ col,
                          const float* __restrict__ h, const float* __restrict__ dinv,
                          float* __restrict__ out, int e_cnt) {
    int idx = blockIdx.x * blockDim.x + threadIdx.x;
    int e = idx >> 4;
    if (e >= e_cnt) return;
    int q4 = (idx & 15) << 2;
    int r = row[e];
    int c = col[e];
    float norm = dinv[r] * dinv[c];
    float4 v = *(const float4*)(h + (size_t)r * OUTF + q4);
    float* op = out + (size_t)c * OUTF + q4;
    atomicAdd(op + 0, v.x * norm);
    atomicAdd(op + 1, v.y * norm);
    atomicAdd(op + 2, v.z * norm);
    atomicAdd(op + 3, v.w * norm);
}

// ---------------- out = relu(out + bias) ----------------
__global__ void k_finish(float* __restrict__ out, const float* __restrict__ bias, int total) {
    int idx = blockIdx.x * blockDim.x + threadIdx.x;
    if (idx >= total) return;
    float v = out[idx] + bias[idx & (OUTF - 1)];
    out[idx] = v > 0.0f ? v : 0.0f;
}

extern "C" void kernel_launch(void* const* d_in, const int* in_sizes, int n_in,
                              void* d_out, int out_size, void* d_ws, size_t ws_size,
                              hipStream_t stream) {
    (void)in_sizes; (void)n_in; (void)out_size; (void)ws_size;

    const float* x    = (const float*)d_in[0];
    const int*   ei   = (const int*)d_in[1];   // [2, NE] int32 (JAX x64 off)
    const float* W    = (const float*)d_in[2];
    const float* bias = (const float*)d_in[3];
    float*       out  = (float*)d_out;

    float* h    = (float*)d_ws;                  // NN*OUTF f32 = 25.6 MB
    float* dinv = h + (size_t)NN * OUTF;         // NN f32 (deg, then rsqrt in place)

    const int* row = ei;        // sources
    const int* col = ei + NE;   // targets

    const int B = 256;

    // degrees
    k_init_deg <<<(NN + B - 1) / B, B, 0, stream>>>(dinv, NN);
    k_accum_deg<<<(NE + B - 1) / B, B, 0, stream>>>(col, dinv, NE);
    k_rsqrt    <<<(NN + B - 1) / B, B, 0, stream>>>(dinv, NN);

    // h = x @ W : one wave per 16 nodes, 8 waves per block
    {
        int waves  = (NN + 15) / 16;
        int blocks = (waves + 7) / 8;
        k_gemm_wmma<<<blocks, B, 0, stream>>>(x, W, h, NN);
    }

    // out = h*dinv^2 (self loops, full overwrite of out)
    k_selfloop<<<((NN * 16) + B - 1) / B, B, 0, stream>>>(h, dinv, out, NN);

    // edge messages (atomic accumulate)
    {
        long long threads = (long long)NE * 16;
        int blocks = (int)((threads + B - 1) / B);
        k_scatter<<<blocks, B, 0, stream>>>(row, col, h, dinv, out, NE);
    }

    // bias + relu
    k_finish<<<((NN * OUTF) + B - 1) / B, B, 0, stream>>>(out, bias, NN * OUTF);
}